// RandomPick_3770981286558
// MI455X (gfx1250) — compile-verified
//
#include <hip/hip_runtime.h>

// Broadcast channel-select: out[b,h,w,c] = (vector[0,c] >= -5) ? ip1 : ip2.
//
// Roofline: worst case 192 MiB of traffic (~8.6 us at 23.3 TB/s), no matrix
// math -> WMMA/TDM inapplicable; this is a pure VMEM streaming kernel.
//
// Key optimization: each lane owns a fixed float4 group of the C=128 channel
// dim (stride % 32 == 0). If its 4 mask bits are uniform -- true for every
// group with overwhelming probability under vector ~ N(0,1), since
// P(x < -5) ~ 2.9e-7 -- the lane selects its source base pointer ONCE and
// streams a single b128 load per element, cutting read traffic in half
// (192 MiB -> 128 MiB, ~5.7 us). Mixed-mask lanes take a correct two-load
// blend fallback under EXEC masking.

typedef float f32x4 __attribute__((ext_vector_type(4)));

__global__ void __launch_bounds__(256)
select_bcast_kernel(const f32x4* __restrict__ ip1,
                    const f32x4* __restrict__ ip2,
                    const float*  __restrict__ vec,   // [B, C]; only row 0 used
                    f32x4* __restrict__ out,
                    int n4)                           // total float4 elements
{
    const int tid    = blockIdx.x * blockDim.x + threadIdx.x;
    const int stride = gridDim.x * blockDim.x;  // launch guarantees stride % 32 == 0

    // C = 128 floats = 32 float4 groups; this lane's group is fixed for the
    // whole grid-stride loop, so the mask (and the uniformity test) is
    // loop-invariant.
    const int c4 = tid & 31;
    const f32x4 v = ((const f32x4*)vec)[c4];
    const bool m0 = v.x >= -5.0f;
    const bool m1 = v.y >= -5.0f;
    const bool m2 = v.z >= -5.0f;
    const bool m3 = v.w >= -5.0f;
    const bool uniform = (m0 == m1) & (m1 == m2) & (m2 == m3);

    if (uniform) {
        // Fast path (expected for all lanes): single-source stream.
        // Per-lane base-pointer select, hoisted out of the loop.
        const f32x4* __restrict__ src = m0 ? ip1 : ip2;
        for (int i = tid; i < n4; i += stride) {
            // gfx1250 global_prefetch_b8, speculative: past-the-end addresses
            // on the last iteration are silently dropped.
            __builtin_prefetch(&src[i + stride], 0, 0);
            // Single-pass data: non-temporal (TH=NT) so we don't churn the
            // 192 MB L2 with never-reused lines.
            f32x4 r = __builtin_nontemporal_load(&src[i]);
            __builtin_nontemporal_store(r, &out[i]);
        }
    } else {
        // Fallback for lanes whose 4-channel group straddles the mask
        // boundary: load both sources, blend per component.
        for (int i = tid; i < n4; i += stride) {
            __builtin_prefetch(&ip1[i + stride], 0, 0);
            __builtin_prefetch(&ip2[i + stride], 0, 0);
            f32x4 a = __builtin_nontemporal_load(&ip1[i]);
            f32x4 b = __builtin_nontemporal_load(&ip2[i]);
            f32x4 r;
            r.x = m0 ? a.x : b.x;
            r.y = m1 ? a.y : b.y;
            r.z = m2 ? a.z : b.z;
            r.w = m3 ? a.w : b.w;
            __builtin_nontemporal_store(r, &out[i]);
        }
    }
}

extern "C" void kernel_launch(void* const* d_in, const int* in_sizes, int n_in,
                              void* d_out, int out_size, void* d_ws, size_t ws_size,
                              hipStream_t stream) {
    const f32x4* ip1 = (const f32x4*)d_in[0];   // [B,H,W,C] f32
    const f32x4* ip2 = (const f32x4*)d_in[1];   // [B,H,W,C] f32
    const float* vec = (const float*)d_in[2];   // [B,C]     f32
    f32x4*       out = (f32x4*)d_out;

    const int n  = in_sizes[0];   // B*H*W*C = 16,777,216
    const int n4 = n / 4;         // 4,194,304 float4 elements

    // 256 threads (8 wave32) per block; up to 4096 blocks -> 1,048,576 lanes,
    // 4 grid-stride iterations each. blockDim is a multiple of 32, keeping the
    // lane -> channel-group mapping loop-invariant for any grid size.
    const int block = 256;
    int grid = (n4 + block - 1) / block;
    if (grid > 4096) grid = 4096;

    select_bcast_kernel<<<grid, block, 0, stream>>>(ip1, ip2, vec, out, n4);
}